// MessagePassingLayer_22686017258178
// MI455X (gfx1250) — compile-verified
//
#include <hip/hip_runtime.h>

// ---------------------------------------------------------------------------
// MessagePassingLayer for MI455X (gfx1250, wave32).
// B=8, P=1024, H=128, RBF=16, cutoff=8, box=30A  => ~8% neighbor density.
// Dense GEMM (h @ msg_w) uses V_WMMA_F32_16X16X4_F32 (exact fp32).
// Attention path exploits sparsity with deterministic ballot compaction.
// ---------------------------------------------------------------------------

#define HIDDEN   128
#define NPART    1024
#define NBATCH   8
#define NRBF     16
#define MAXN     256          // max compacted neighbors per row (mean ~85)
#define NROWS    (NBATCH * NPART)   // 8192

typedef float v2f __attribute__((ext_vector_type(2)));
typedef float v8f __attribute__((ext_vector_type(8)));

__device__ __forceinline__ float gelu_tanh(float x) {
    // jax.nn.gelu(approximate=True): 0.5*x*(1+tanh(sqrt(2/pi)*(x+0.044715*x^3)))
    float x3 = x * x * x;
    return 0.5f * x * (1.0f + tanhf(0.7978845608028654f * (x + 0.044715f * x3)));
}

// ---------------------------------------------------------------------------
// Kernel 1: transformed = gelu(h @ msg_w + msg_b)     [8192,128]x[128,128]
// One wave per 16x16 output tile, K-loop in steps of 4 via WMMA f32 16x16x4.
// Grid: 512 blocks x 256 threads (8 waves); blockIdx = M-tile, wave = N-tile.
//
// A (16x4 f32) per-lane layout (ISA 7.12.2): lanes 0-15 -> M=lane, K={k0,k0+1};
// lanes 16-31 -> M=lane-16, K={k0+2,k0+3}.  B (4x16) mirrors: lanes 0-15 hold
// first half of K range, lanes 16-31 the second half; N = lane&15.
// C/D (16x16 f32): VGPR r: lanes 0-15 -> M=r, lanes 16-31 -> M=r+8; N=lane&15.
// ---------------------------------------------------------------------------
__global__ __launch_bounds__(256)
void k_transform(const float* __restrict__ h, const float* __restrict__ W,
                 const float* __restrict__ bias, float* __restrict__ out) {
    const int lane   = threadIdx.x & 31;
    const int wave   = threadIdx.x >> 5;      // N-tile 0..7
    const int m_tile = blockIdx.x;            // 0..511
    const int half   = lane >> 4;             // 0 or 1
    const int mn     = lane & 15;
    const int row    = m_tile * 16 + mn;      // A row for this lane
    const int col    = wave * 16 + mn;        // B/C column for this lane
    const int khalf  = half * 2;

    v8f c;
#pragma unroll
    for (int i = 0; i < 8; ++i) c[i] = 0.0f;

    const float* hrow = h + (size_t)row * HIDDEN;
#pragma unroll 4
    for (int k0 = 0; k0 < HIDDEN; k0 += 4) {
        const int kb = k0 + khalf;
        v2f a = *(const v2f*)(hrow + kb);                 // {A[row][kb], A[row][kb+1]}
        v2f b;
        b.x = W[(size_t)kb * HIDDEN + col];               // {W[kb][col], W[kb+1][col]}
        b.y = W[(size_t)(kb + 1) * HIDDEN + col];
        c = __builtin_amdgcn_wmma_f32_16x16x4_f32(
                /*neg_a=*/false, a, /*neg_b=*/false, b,
                /*c_mod=*/(short)0, c, /*reuse_a=*/false, /*reuse_b=*/false);
    }

    const float bn    = bias[col];
    const int   mbase = m_tile * 16 + half * 8;
#pragma unroll
    for (int r = 0; r < 8; ++r) {
        float x = c[r] + bn;
        out[(size_t)(mbase + r) * HIDDEN + col] = gelu_tanh(x);
    }
}

// ---------------------------------------------------------------------------
// Kernel 2: deterministic neighbor compaction. One wave per (b,p) row.
// Ballot + popcount prefix keeps entries in ascending-j order => bitwise
// deterministic softmax/aggregation later. Grid: 1024 blocks x 256 threads.
// ---------------------------------------------------------------------------
__global__ __launch_bounds__(256)
void k_neighbors(const float* __restrict__ pos, const unsigned char* __restrict__ mask,
                 int* __restrict__ cnt, int* __restrict__ idx, float* __restrict__ dst) {
    const int lane = threadIdx.x & 31;
    const int wid  = blockIdx.x * 8 + (threadIdx.x >> 5);   // 0..8191
    const int b    = wid >> 10;
    const int p    = wid & (NPART - 1);

    const float* pb = pos + (size_t)b * NPART * 3;
    const float xi = pb[p * 3 + 0];
    const float yi = pb[p * 3 + 1];
    const float zi = pb[p * 3 + 2];
    const bool  mi = mask[b * NPART + p] != 0;

    const int base = wid * MAXN;
    int c = 0;
    for (int ch = 0; ch < NPART; ch += 32) {
        const int j = ch + lane;
        const float dx = pb[j * 3 + 0] - xi;
        const float dy = pb[j * 3 + 1] - yi;
        const float dz = pb[j * 3 + 2] - zi;
        const float d  = sqrtf(dx * dx + dy * dy + dz * dz);
        const bool  mj = mask[b * NPART + j] != 0;
        const bool  pred = mi && mj && (d < 8.0f) && (d > 0.01f);
        const unsigned int bal = __builtin_amdgcn_ballot_w32(pred);
        if (pred) {
            const int slot = c + __popc(bal & ((1u << lane) - 1u));
            if (slot < MAXN) { idx[base + slot] = j; dst[base + slot] = d; }
        }
        c += __popc(bal);
    }
    if (lane == 0) cnt[wid] = (c < MAXN) ? c : MAXN;
}

// ---------------------------------------------------------------------------
// Kernel 3: attention scores (RBF -> 16x32x1 gelu MLP) + row softmax, only on
// compacted neighbors. One block (256 thr) per row; waves beyond cnt skip the
// MLP entirely (compaction payoff). Grid: 8192 blocks.
// Non-neighbor logits are -60000 in the reference: exp(-60000-max) == 0.0f in
// fp32, so restricting the softmax to neighbors is exact.
// ---------------------------------------------------------------------------
__global__ __launch_bounds__(256)
void k_score_softmax(const int* __restrict__ cnt, const float* __restrict__ dst,
                     const float* __restrict__ aw1, const float* __restrict__ ab1,
                     const float* __restrict__ aw2, const float* __restrict__ ab2,
                     float* __restrict__ wgt) {
    __shared__ float s_aw1[NRBF * 32];
    __shared__ float s_ab1[32];
    __shared__ float s_aw2[32];
    __shared__ float s_ab2;
    __shared__ float red[256];

    const int tid = threadIdx.x;
    for (int t = tid; t < NRBF * 32; t += 256) s_aw1[t] = aw1[t];
    if (tid < 32) { s_ab1[tid] = ab1[tid]; s_aw2[tid] = aw2[tid]; }
    if (tid == 0) s_ab2 = ab2[0];
    __syncthreads();

    const int row = blockIdx.x;
    const int n   = cnt[row];
    const bool act = tid < n;

    float sc = 0.0f;
    if (act) {
        const float d = dst[row * MAXN + tid];
        // Gaussian RBF, centers = linspace(0, 8, 16), gamma = 2.
        // Direct expf per bin (a ratio-iteration underflows: exp(-2*d^2) can
        // drop below FLT_MIN while high bins are ~1).
        float rbf[NRBF];
        const float delta = 8.0f / 15.0f;
#pragma unroll
        for (int k = 0; k < NRBF; ++k) {
            const float t = d - (float)k * delta;
            rbf[k] = expf(-2.0f * t * t);
        }
        float hid[32];
#pragma unroll
        for (int o = 0; o < 32; ++o) hid[o] = s_ab1[o];
#pragma unroll
        for (int k = 0; k < NRBF; ++k) {
            const float rv = rbf[k];
#pragma unroll
            for (int o = 0; o < 32; ++o) hid[o] = fmaf(rv, s_aw1[k * 32 + o], hid[o]);
        }
        float acc = s_ab2;
#pragma unroll
        for (int o = 0; o < 32; ++o) acc = fmaf(gelu_tanh(hid[o]), s_aw2[o], acc);
        sc = acc;
    }

    // block softmax over the n compacted entries
    red[tid] = act ? sc : -3.402823466e+38f;
    __syncthreads();
    for (int s = 128; s > 0; s >>= 1) {
        if (tid < s) red[tid] = fmaxf(red[tid], red[tid + s]);
        __syncthreads();
    }
    const float m = red[0];
    __syncthreads();
    const float e = act ? expf(sc - m) : 0.0f;
    red[tid] = e;
    __syncthreads();
    for (int s = 128; s > 0; s >>= 1) {
        if (tid < s) red[tid] += red[tid + s];
        __syncthreads();
    }
    const float sum = red[0];
    if (act) wgt[row * MAXN + tid] = e / sum;
}

// ---------------------------------------------------------------------------
// Kernel 4: msg = sparse(attn) @ transformed, residual, LayerNorm.
// One block (128 thr = one column each) per row. Neighbor (j, w) pairs are
// staged in LDS and broadcast-read. cnt==0 edge case (all logits -60000 =>
// uniform softmax 1/1024) handled exactly. Grid: 8192 blocks.
// ---------------------------------------------------------------------------
__global__ __launch_bounds__(128)
void k_agg_ln(const float* __restrict__ h, const float* __restrict__ tr,
              const int* __restrict__ cnt, const int* __restrict__ idx,
              const float* __restrict__ wgt, const float* __restrict__ ln_g,
              const float* __restrict__ ln_b, float* __restrict__ out) {
    __shared__ float sw[MAXN];
    __shared__ int   sj[MAXN];
    __shared__ float red[128];

    const int tid = threadIdx.x;       // column 0..127
    const int row = blockIdx.x;        // 0..8191
    const int b   = row >> 10;
    const int n   = cnt[row];

    for (int t = tid; t < n; t += 128) {
        sj[t] = idx[row * MAXN + t];
        sw[t] = wgt[row * MAXN + t];
    }
    __syncthreads();

    const float* trb = tr + (size_t)b * NPART * HIDDEN;
    float acc = 0.0f;
    if (n > 0) {
        for (int e = 0; e < n; ++e)
            acc = fmaf(sw[e], trb[(size_t)sj[e] * HIDDEN + tid], acc);
    } else {
        for (int j = 0; j < NPART; ++j) acc += trb[(size_t)j * HIDDEN + tid];
        acc *= (1.0f / (float)NPART);
    }

    const float x = h[(size_t)row * HIDDEN + tid] + acc;

    // LayerNorm over 128 columns
    red[tid] = x;
    __syncthreads();
    for (int s = 64; s > 0; s >>= 1) {
        if (tid < s) red[tid] += red[tid + s];
        __syncthreads();
    }
    const float mu = red[0] * (1.0f / (float)HIDDEN);
    __syncthreads();
    const float dx = x - mu;
    red[tid] = dx * dx;
    __syncthreads();
    for (int s = 64; s > 0; s >>= 1) {
        if (tid < s) red[tid] += red[tid + s];
        __syncthreads();
    }
    const float var = red[0] * (1.0f / (float)HIDDEN);
    out[(size_t)row * HIDDEN + tid] =
        dx * (1.0f / sqrtf(var + 1e-5f)) * ln_g[tid] + ln_b[tid];
}

// ---------------------------------------------------------------------------
// Host launcher. Scratch layout in d_ws (bytes):
//   [0,           32768)  cnt   : 8192 int
//   [32768,     8421376)  idx   : 8192*256 int
//   [8421376,  16809984)  dist  : 8192*256 float
//   [16809984, 25198592)  wgt   : 8192*256 float
//   [25198592, 29392896)  transformed : 8192*128 float   (~28 MB total)
// ---------------------------------------------------------------------------
extern "C" void kernel_launch(void* const* d_in, const int* in_sizes, int n_in,
                              void* d_out, int out_size, void* d_ws, size_t ws_size,
                              hipStream_t stream) {
    (void)in_sizes; (void)n_in; (void)out_size; (void)ws_size;

    const float*         h     = (const float*)d_in[0];
    const float*         pos   = (const float*)d_in[1];
    const unsigned char* mask  = (const unsigned char*)d_in[2]; // jax bool = 1 byte
    const float*         msg_w = (const float*)d_in[3];
    const float*         msg_b = (const float*)d_in[4];
    const float*         aw1   = (const float*)d_in[5];
    const float*         ab1   = (const float*)d_in[6];
    const float*         aw2   = (const float*)d_in[7];
    const float*         ab2   = (const float*)d_in[8];
    const float*         ln_g  = (const float*)d_in[9];
    const float*         ln_b  = (const float*)d_in[10];
    float*               out   = (float*)d_out;

    char* ws = (char*)d_ws;
    int*   cnt  = (int*)(ws);
    int*   idx  = (int*)(ws + 32768);
    float* dist = (float*)(ws + 32768 + (size_t)NROWS * MAXN * 4);
    float* wgt  = (float*)(ws + 32768 + 2 * (size_t)NROWS * MAXN * 4);
    float* tr   = (float*)(ws + 32768 + 3 * (size_t)NROWS * MAXN * 4);

    // 1) transformed = gelu(h @ msg_w + msg_b)   (WMMA f32 16x16x4)
    k_transform<<<NROWS / 16, 256, 0, stream>>>(h, msg_w, msg_b, tr);
    // 2) deterministic neighbor compaction
    k_neighbors<<<NROWS / 8, 256, 0, stream>>>(pos, mask, cnt, idx, dist);
    // 3) per-pair MLP scores + row softmax over compacted neighbors
    k_score_softmax<<<NROWS, 256, 0, stream>>>(cnt, dist, aw1, ab1, aw2, ab2, wgt);
    // 4) sparse aggregation + residual + LayerNorm
    k_agg_ln<<<NROWS, 128, 0, stream>>>(h, tr, cnt, idx, wgt, ln_g, ln_b, out);
}